// SphConv3_62904091017705
// MI455X (gfx1250) — compile-verified
//
#include <hip/hip_runtime.h>
#include <hip/hip_bf16.h>

typedef __attribute__((ext_vector_type(2))) float v2f;
typedef __attribute__((ext_vector_type(8))) float v8f;

#define LN_EPS 1e-5f

__device__ __forceinline__ v8f wmma_f32_k4(v2f a, v2f b, v8f c) {
    // D = A(16x4,f32) * B(4x16,f32) + C(16x16,f32)
    return __builtin_amdgcn_wmma_f32_16x16x4_f32(
        /*neg_a=*/false, a, /*neg_b=*/false, b,
        /*c_mod=*/(short)0, c, /*reuse_a=*/false, /*reuse_b=*/false);
}

__device__ __forceinline__ void atomic_add_f32(float* p, float v) {
    __hip_atomic_fetch_add(p, v, __ATOMIC_RELAXED, __HIP_MEMORY_SCOPE_AGENT);
}

// Async direct memory->LDS copy (GLOBAL_LOAD_ASYNC_TO_LDS_B128, ASYNCcnt).
// lds_addr: LDS byte offset (low 32 bits of generic shared pointer).
__device__ __forceinline__ void async_load_b128_to_lds(unsigned lds_addr,
                                                       const void* gaddr) {
    asm volatile("global_load_async_to_lds_b128 %0, %1, off"
                 :: "v"(lds_addr), "v"(gaddr) : "memory");
}
__device__ __forceinline__ void wait_asynccnt0() {
    asm volatile("s_wait_asynccnt 0" ::: "memory");
}

// ---------------------------------------------------------------------------
// Zero the aggregation buffers (graph-replay safe: every call starts clean)
// ---------------------------------------------------------------------------
__global__ void zero_ws_kernel(float* __restrict__ p, size_t n) {
    size_t idx = (size_t)blockIdx.x * blockDim.x + threadIdx.x;
    size_t stride = (size_t)gridDim.x * blockDim.x;
    for (size_t t = idx; t < n; t += stride) p[t] = 0.0f;
}

// ---------------------------------------------------------------------------
// _x = x @ W_down + b_down        [N,256] @ [256,64] -> [N,64]
// wave-per-16x16-tile fp32 WMMA; 4 waves per block cover all 64 columns
// ---------------------------------------------------------------------------
__global__ void down_proj_kernel(const float* __restrict__ x,
                                 const float* __restrict__ Wd,
                                 const float* __restrict__ bd,
                                 float* __restrict__ xp, int N) {
    const int D = 256, P = 64;
    const int row0 = blockIdx.x * 16;
    const int wave = threadIdx.x >> 5;      // 0..3 -> column tile
    const int lane = threadIdx.x & 31;
    const int lm = lane & 15;
    const int half = lane >> 4;             // 0: K pair {0,1}, 1: {2,3}
    const int col0 = wave * 16;

    int ra = row0 + lm;
    if (ra >= N) ra = N - 1;                // clamp: stores are guarded

    v8f acc = {};
    for (int k = 0; k < D; k += 4) {
        const int ka = k + half * 2;
        v2f a, b;
        a.x = x[(size_t)ra * D + ka];
        a.y = x[(size_t)ra * D + ka + 1];
        b.x = Wd[(size_t)ka * P + col0 + lm];
        b.y = Wd[(size_t)(ka + 1) * P + col0 + lm];
        acc = wmma_f32_k4(a, b, acc);
    }
    const float bias = bd[col0 + lm];
    if (row0 + 16 <= N) {                   // fast path: full tile (uniform)
        float* dst = xp + (size_t)(row0 + half * 8) * P + col0 + lm;
        #pragma unroll
        for (int r = 0; r < 8; ++r) dst[(size_t)r * P] = acc[r] + bias;
    } else {
        for (int r = 0; r < 8; ++r) {
            const int row = row0 + r + half * 8;
            if (row < N) xp[(size_t)row * P + col0 + lm] = acc[r] + bias;
        }
    }
}

// ---------------------------------------------------------------------------
// Edge kernel: l0_pair = _x[j] * rbf * factor; scatter into agg0/agg1/aggh
// one wave per edge; lane handles 2 consecutive P elements (float2)
// ---------------------------------------------------------------------------
__global__ void edge_scatter_kernel(const float* __restrict__ xp,
                                    const float* __restrict__ rbf,
                                    const float* __restrict__ factor,
                                    const float* __restrict__ sph0,  // [E,5]
                                    const float* __restrict__ sph1,  // [E,3]
                                    const int* __restrict__ jidx,
                                    const int* __restrict__ iidx,
                                    float* __restrict__ agg0,        // [N,64]
                                    float* __restrict__ agg1,        // [N,3,64]
                                    float* __restrict__ aggh,        // [N,5,64]
                                    int E) {
    const int P = 64;
    const int wavesPerBlk = blockDim.x >> 5;
    const int e = blockIdx.x * wavesPerBlk + (threadIdx.x >> 5);
    if (e >= E) return;
    const int lane = threadIdx.x & 31;

    // prefetch next edge's rbf row into cache (global_prefetch)
    if (e + wavesPerBlk < E)
        __builtin_prefetch(rbf + (size_t)(e + wavesPerBlk) * P + lane * 2, 0, 1);

    const int js = jidx[e];
    const int is = iidx[e];
    const float f = factor[e];

    const v2f* xp2  = reinterpret_cast<const v2f*>(xp);
    const v2f* rbf2 = reinterpret_cast<const v2f*>(rbf);

    const v2f v = xp2[(size_t)js * (P / 2) + lane];  // L2-resident gather
    // rbf is streamed exactly once: non-temporal so it doesn't evict the
    // _x gather table / aggregation lines from L2
    const v2f r = __builtin_nontemporal_load(rbf2 + (size_t)e * (P / 2) + lane);

    const float l0x = v.x * r.x * f;
    const float l0y = v.y * r.y * f;
    const int p = lane * 2;

    atomic_add_f32(&agg0[(size_t)is * P + p],     l0x);
    atomic_add_f32(&agg0[(size_t)is * P + p + 1], l0y);

    #pragma unroll
    for (int c = 0; c < 3; ++c) {
        const float s = sph1[(size_t)e * 3 + c];
        float* dst = &agg1[((size_t)is * 3 + c) * P + p];
        atomic_add_f32(dst,     l0x * s);
        atomic_add_f32(dst + 1, l0y * s);
    }
    #pragma unroll
    for (int c = 0; c < 5; ++c) {
        const float s = sph0[(size_t)e * 5 + c];
        float* dst = &aggh[((size_t)is * 5 + c) * P + p];
        atomic_add_f32(dst,     l0x * s);
        atomic_add_f32(dst + 1, l0y * s);
    }
}

// ---------------------------------------------------------------------------
// Up-projection: node[N,384] =
//   [ (_x*agg0)@W_up0 + b_up0 | sum_c sq((_x*agg1_c)@W_up1) | sum_c sq((_x*aggh_c)@W_uph) ]
// A-matrices (element products) staged in LDS; 8 waves = 8 column tiles of 16
// ---------------------------------------------------------------------------
__global__ void up_proj_kernel(const float* __restrict__ xp,
                               const float* __restrict__ agg0,
                               const float* __restrict__ agg1,
                               const float* __restrict__ aggh,
                               const float* __restrict__ Wu0,  // [64,128]
                               const float* __restrict__ bu0,  // [128]
                               const float* __restrict__ Wu1,  // [64,128]
                               const float* __restrict__ Wuh,  // [64,128]
                               float* __restrict__ node,       // [N,384]
                               int N) {
    const int P = 64, TP = 128, NODE_W = 384;
    __shared__ float sA[9 * 16 * 64];   // 36 KB: ch0=agg0, ch1..3=agg1, ch4..8=aggh

    const int row0 = blockIdx.x * 16;
    const bool full = (row0 + 16 <= N);

    // cooperative build of all nine 16x64 A-matrices (element products)
    for (int t = threadIdx.x; t < 9 * 16 * 64; t += blockDim.x) {
        const int ch = t >> 10;          // /1024
        const int rem = t & 1023;
        const int m = rem >> 6;
        const int k = rem & 63;
        int r = row0 + m;
        if (r >= N) r = N - 1;
        const float xv = xp[(size_t)r * P + k];
        float av;
        if (ch == 0)      av = agg0[(size_t)r * P + k];
        else if (ch < 4)  av = agg1[((size_t)r * 3 + (ch - 1)) * P + k];
        else              av = aggh[((size_t)r * 5 + (ch - 4)) * P + k];
        sA[t] = xv * av;
    }
    __syncthreads();

    const int wave = threadIdx.x >> 5;  // 0..7 -> column tile of the 128 cols
    const int lane = threadIdx.x & 31;
    const int lm = lane & 15;
    const int half = lane >> 4;
    const int col0 = wave * 16;

    // ---- l0 section: 1 GEMM, K=64 ----
    {
        const float* A = sA;  // ch 0
        v8f acc = {};
        for (int k = 0; k < P; k += 4) {
            const int ka = k + half * 2;
            v2f a, b;
            a.x = A[lm * 64 + ka];
            a.y = A[lm * 64 + ka + 1];
            b.x = Wu0[(size_t)ka * TP + col0 + lm];
            b.y = Wu0[(size_t)(ka + 1) * TP + col0 + lm];
            acc = wmma_f32_k4(a, b, acc);
        }
        const float bias = bu0[col0 + lm];
        if (full) {
            float* dst = node + (size_t)(row0 + half * 8) * NODE_W + col0 + lm;
            #pragma unroll
            for (int r = 0; r < 8; ++r) dst[(size_t)r * NODE_W] = acc[r] + bias;
        } else {
            for (int r = 0; r < 8; ++r) {
                const int row = row0 + r + half * 8;
                if (row < N) node[(size_t)row * NODE_W + col0 + lm] = acc[r] + bias;
            }
        }
    }

    // ---- l1 section: 3 GEMMs, squared and channel-summed ----
    {
        v8f sum = {};
        for (int c = 0; c < 3; ++c) {
            const float* A = sA + (1 + c) * 1024;
            v8f acc = {};
            for (int k = 0; k < P; k += 4) {
                const int ka = k + half * 2;
                v2f a, b;
                a.x = A[lm * 64 + ka];
                a.y = A[lm * 64 + ka + 1];
                b.x = Wu1[(size_t)ka * TP + col0 + lm];
                b.y = Wu1[(size_t)(ka + 1) * TP + col0 + lm];
                acc = wmma_f32_k4(a, b, acc);
            }
            for (int r = 0; r < 8; ++r) sum[r] += acc[r] * acc[r];
        }
        if (full) {
            float* dst = node + (size_t)(row0 + half * 8) * NODE_W + 128 + col0 + lm;
            #pragma unroll
            for (int r = 0; r < 8; ++r) dst[(size_t)r * NODE_W] = sum[r];
        } else {
            for (int r = 0; r < 8; ++r) {
                const int row = row0 + r + half * 8;
                if (row < N) node[(size_t)row * NODE_W + 128 + col0 + lm] = sum[r];
            }
        }
    }

    // ---- lh section: 5 GEMMs, squared and channel-summed ----
    {
        v8f sum = {};
        for (int c = 0; c < 5; ++c) {
            const float* A = sA + (4 + c) * 1024;
            v8f acc = {};
            for (int k = 0; k < P; k += 4) {
                const int ka = k + half * 2;
                v2f a, b;
                a.x = A[lm * 64 + ka];
                a.y = A[lm * 64 + ka + 1];
                b.x = Wuh[(size_t)ka * TP + col0 + lm];
                b.y = Wuh[(size_t)(ka + 1) * TP + col0 + lm];
                acc = wmma_f32_k4(a, b, acc);
            }
            for (int r = 0; r < 8; ++r) sum[r] += acc[r] * acc[r];
        }
        if (full) {
            float* dst = node + (size_t)(row0 + half * 8) * NODE_W + 256 + col0 + lm;
            #pragma unroll
            for (int r = 0; r < 8; ++r) dst[(size_t)r * NODE_W] = sum[r];
        } else {
            for (int r = 0; r < 8; ++r) {
                const int row = row0 + r + half * 8;
                if (row < N) node[(size_t)row * NODE_W + 256 + col0 + lm] = sum[r];
            }
        }
    }
}

// ---------------------------------------------------------------------------
// h = silu(layernorm(node @ W_n1 + b_n1))    [N,384]@[384,256] + fused LN/SiLU
// row block staged via async memory->LDS copy (ASYNCcnt)
// ---------------------------------------------------------------------------
__global__ void node_mlp1_kernel(const float* __restrict__ node,
                                 const float* __restrict__ Wn1,  // [384,256]
                                 const float* __restrict__ bn1,
                                 const float* __restrict__ lng,
                                 const float* __restrict__ lnb,
                                 float* __restrict__ h, int N) {
    const int K = 384, D = 256;
    __shared__ float sN[16 * 384];   // 24 KB input row block
    __shared__ float sO[16 * 256];   // 16 KB output tile (pre-LN)
    __shared__ float sMean[16], sRstd[16];

    const int row0 = blockIdx.x * 16;

    // async stage: 16 rows x 384 cols, 16B granules straight into LDS
    {
        const unsigned sN_base = (unsigned)(uintptr_t)&sN[0];
        for (int t = threadIdx.x; t < 16 * (K / 4); t += blockDim.x) {
            const int m = t / (K / 4);
            const int k4 = t % (K / 4);
            int r = row0 + m;
            if (r >= N) r = N - 1;
            async_load_b128_to_lds(sN_base + (unsigned)(m * K + k4 * 4) * 4u,
                                   node + (size_t)r * K + k4 * 4);
        }
        wait_asynccnt0();
    }
    __syncthreads();

    const int wave = threadIdx.x >> 5;
    const int lane = threadIdx.x & 31;
    const int lm = lane & 15;
    const int half = lane >> 4;

    for (int ct = wave; ct < 16; ct += 8) {       // 8 waves x 2 column tiles
        const int col0 = ct * 16;
        v8f acc = {};
        for (int k = 0; k < K; k += 4) {
            const int ka = k + half * 2;
            v2f a, b;
            a.x = sN[lm * K + ka];
            a.y = sN[lm * K + ka + 1];
            b.x = Wn1[(size_t)ka * D + col0 + lm];
            b.y = Wn1[(size_t)(ka + 1) * D + col0 + lm];
            acc = wmma_f32_k4(a, b, acc);
        }
        const float bias = bn1[col0 + lm];
        #pragma unroll
        for (int r = 0; r < 8; ++r)
            sO[(r + half * 8) * D + col0 + lm] = acc[r] + bias;
    }
    __syncthreads();

    // per-row LayerNorm statistics
    if (threadIdx.x < 16) {
        float s = 0.f, s2 = 0.f;
        const float* row = &sO[threadIdx.x * D];
        for (int c = 0; c < D; ++c) { const float v = row[c]; s += v; s2 += v * v; }
        const float m = s * (1.0f / D);
        const float var = s2 * (1.0f / D) - m * m;
        sMean[threadIdx.x] = m;
        sRstd[threadIdx.x] = rsqrtf(var + LN_EPS);
    }
    __syncthreads();

    if (row0 + 16 <= N) {                          // fast path (uniform)
        for (int t = threadIdx.x; t < 16 * D; t += blockDim.x) {
            const int m = t / D, c = t % D;
            const float v = (sO[t] - sMean[m]) * sRstd[m] * lng[c] + lnb[c];
            const float sig = 1.0f / (1.0f + __expf(-v));
            h[(size_t)(row0 + m) * D + c] = v * sig;
        }
    } else {
        for (int t = threadIdx.x; t < 16 * D; t += blockDim.x) {
            const int m = t / D, c = t % D;
            const int r = row0 + m;
            if (r < N) {
                const float v = (sO[t] - sMean[m]) * sRstd[m] * lng[c] + lnb[c];
                const float sig = 1.0f / (1.0f + __expf(-v));
                h[(size_t)r * D + c] = v * sig;
            }
        }
    }
}

// ---------------------------------------------------------------------------
// out = h @ W_n2 + b_n2 + x      [N,256]@[256,256] + bias + residual
// 16 waves/block; input row block staged via async memory->LDS copy
// ---------------------------------------------------------------------------
__global__ void node_mlp2_kernel(const float* __restrict__ h,
                                 const float* __restrict__ Wn2,
                                 const float* __restrict__ bn2,
                                 const float* __restrict__ x,
                                 float* __restrict__ out, int N) {
    const int D = 256;
    __shared__ float sH[16 * 256];   // 16 KB

    const int row0 = blockIdx.x * 16;

    {
        const unsigned sH_base = (unsigned)(uintptr_t)&sH[0];
        for (int t = threadIdx.x; t < 16 * (D / 4); t += blockDim.x) {
            const int m = t / (D / 4);
            const int k4 = t % (D / 4);
            int r = row0 + m;
            if (r >= N) r = N - 1;
            async_load_b128_to_lds(sH_base + (unsigned)(m * D + k4 * 4) * 4u,
                                   h + (size_t)r * D + k4 * 4);
        }
        wait_asynccnt0();
    }
    __syncthreads();

    const int wave = threadIdx.x >> 5;      // 0..15 -> column tile
    const int lane = threadIdx.x & 31;
    const int lm = lane & 15;
    const int half = lane >> 4;
    const int col0 = wave * 16;

    v8f acc = {};
    for (int k = 0; k < D; k += 4) {
        const int ka = k + half * 2;
        v2f a, b;
        a.x = sH[lm * D + ka];
        a.y = sH[lm * D + ka + 1];
        b.x = Wn2[(size_t)ka * D + col0 + lm];
        b.y = Wn2[(size_t)(ka + 1) * D + col0 + lm];
        acc = wmma_f32_k4(a, b, acc);
    }
    const float bias = bn2[col0 + lm];
    if (row0 + 16 <= N) {                   // fast path: full tile (uniform)
        const size_t base = (size_t)(row0 + half * 8) * D + col0 + lm;
        #pragma unroll
        for (int r = 0; r < 8; ++r)
            out[base + (size_t)r * D] = acc[r] + bias + x[base + (size_t)r * D];
    } else {
        for (int r = 0; r < 8; ++r) {
            const int row = row0 + r + half * 8;
            if (row < N)
                out[(size_t)row * D + col0 + lm] =
                    acc[r] + bias + x[(size_t)row * D + col0 + lm];
        }
    }
}

// ---------------------------------------------------------------------------
extern "C" void kernel_launch(void* const* d_in, const int* in_sizes, int n_in,
                              void* d_out, int out_size, void* d_ws, size_t ws_size,
                              hipStream_t stream) {
    const float* x      = (const float*)d_in[0];
    const float* rbf    = (const float*)d_in[1];
    const float* factor = (const float*)d_in[2];
    const float* sph0   = (const float*)d_in[3];   // [E,5,1]
    const float* sph1   = (const float*)d_in[4];   // [E,3,1]
    const float* Wd     = (const float*)d_in[5];
    const float* bd     = (const float*)d_in[6];
    const float* Wu0    = (const float*)d_in[7];
    const float* bu0    = (const float*)d_in[8];
    const float* Wu1    = (const float*)d_in[9];
    const float* Wuh    = (const float*)d_in[10];
    const float* Wn1    = (const float*)d_in[11];
    const float* bn1    = (const float*)d_in[12];
    const float* lng    = (const float*)d_in[13];
    const float* lnb    = (const float*)d_in[14];
    const float* Wn2    = (const float*)d_in[15];
    const float* bn2    = (const float*)d_in[16];
    const int*   jj     = (const int*)d_in[17];
    const int*   ii     = (const int*)d_in[18];

    const int N = in_sizes[0] / 256;     // 25000
    const int E = in_sizes[17];          // 500000

    // workspace layout (floats): xp | agg0 | agg1 | aggh | node | h
    float* ws   = (float*)d_ws;
    float* xp   = ws;
    float* agg0 = xp   + (size_t)N * 64;
    float* agg1 = agg0 + (size_t)N * 64;
    float* aggh = agg1 + (size_t)N * 192;
    float* node = aggh + (size_t)N * 320;
    float* hbuf = node + (size_t)N * 384;
    float* out  = (float*)d_out;

    const int nrb = (N + 15) / 16;       // node row blocks

    // 1) zero aggregation buffers (agg0+agg1+aggh = N*576 floats)
    zero_ws_kernel<<<2048, 256, 0, stream>>>(agg0, (size_t)N * 576);

    // 2) _x = x @ W_down + b_down  (WMMA f32)
    down_proj_kernel<<<nrb, 128, 0, stream>>>(x, Wd, bd, xp, N);

    // 3) edge gather/scale + atomic scatter into 9 aggregation channels
    edge_scatter_kernel<<<(E + 7) / 8, 256, 0, stream>>>(
        xp, rbf, factor, sph0, sph1, jj, ii, agg0, agg1, aggh, E);

    // 4) nine fused up-projection GEMMs -> node[N,384]  (WMMA f32)
    up_proj_kernel<<<nrb, 256, 0, stream>>>(
        xp, agg0, agg1, aggh, Wu0, bu0, Wu1, Wuh, node, N);

    // 5) node @ W_n1 + LayerNorm + SiLU  (WMMA f32, async-LDS staging)
    node_mlp1_kernel<<<nrb, 256, 0, stream>>>(node, Wn1, bn1, lng, lnb, hbuf, N);

    // 6) h @ W_n2 + b_n2 + x  (WMMA f32, async-LDS staging, fused residual)
    node_mlp2_kernel<<<nrb, 512, 0, stream>>>(hbuf, Wn2, bn2, x, out, N);
}